// TurboQuantValue_73177652789666
// MI455X (gfx1250) — compile-verified
//
#include <hip/hip_runtime.h>
#include <stdint.h>

// ---------------------------------------------------------------------------
// TurboQuantValue forward (quantize + dequantize), CDNA5 / gfx1250.
//
// x: [4,32,4096,128] f32 -> out same shape. One 128-float row per wave32.
// Data path: global -> (async tensor-unit copy) -> LDS -> VGPR, double
// buffered per wave, tracked with ASYNCcnt. Output written with nontemporal
// b128 stores (write-once stream, don't pollute L2).
// ---------------------------------------------------------------------------

typedef __attribute__((ext_vector_type(4))) float v4f;

#define WAVES_PER_BLOCK 8
#define GROUP 128

// Converged Lloyd-Max codebook for unit Laplace (b=1) on [-10,10], 16 levels.
// Symmetric: levels = +/-P[k]; decision boundaries = midpoints PB[k].
#define P0 0.17409f
#define P1 0.56807f
#define P2 1.02187f
#define P3 1.55679f
#define P4 2.20871f
#define P5 3.04269f
#define P6 4.20239f
#define P7 6.12500f
#define PB0 0.37108f   // (P0+P1)/2
#define PB1 0.79497f
#define PB2 1.28933f
#define PB3 1.88275f
#define PB4 2.62570f
#define PB5 3.62254f
#define PB6 5.16370f

__device__ __forceinline__ float lm_dequant(float u) {
  // searchsorted(side='left') with symmetric boundaries: strict '>' compares,
  // sign applied last. u==0 (the row min) falls on boundary b7=0 and must map
  // to the NEGATIVE first level -> (u > 0) ? p : -p.
  float a = fabsf(u);
  float p = P0;
  p = (a > PB0) ? P1 : p;
  p = (a > PB1) ? P2 : p;
  p = (a > PB2) ? P3 : p;
  p = (a > PB3) ? P4 : p;
  p = (a > PB4) ? P5 : p;
  p = (a > PB5) ? P6 : p;
  p = (a > PB6) ? P7 : p;
  return (u > 0.0f) ? p : -p;
}

__device__ __forceinline__ float wave_min(float v) {
#pragma unroll
  for (int off = 16; off; off >>= 1) v = fminf(v, __shfl_xor(v, off, 32));
  return v;
}
__device__ __forceinline__ float wave_sum(float v) {
#pragma unroll
  for (int off = 16; off; off >>= 1) v += __shfl_xor(v, off, 32);
  return v;
}

// Async tensor-unit copy: 16B per lane, global -> LDS. GV mode: vdst = per-lane
// LDS byte offset, vaddr = per-lane 64-bit global address. Tracked by ASYNCcnt.
__device__ __forceinline__ void async_row_to_lds(const float* gsrc, uint32_t lds_off) {
  asm volatile("global_load_async_to_lds_b128 %0, %1, off"
               :: "v"(lds_off), "v"(gsrc)
               : "memory");
}

__global__ __launch_bounds__(256)
void turboquant_fwd(const float* __restrict__ x, float* __restrict__ out, int nrows) {
  __shared__ float smem[WAVES_PER_BLOCK][2][GROUP];   // 8 KB: 2 row buffers/wave

  const int lane = threadIdx.x & 31;
  const int wv   = threadIdx.x >> 5;
  const int gw   = blockIdx.x * WAVES_PER_BLOCK + wv;   // global wave id
  const int nw   = gridDim.x * WAVES_PER_BLOCK;         // total waves

  int row = gw;
  if (row >= nrows) return;                             // wave-uniform exit

  // Per-lane LDS byte offsets (flat-address low 32 bits == LDS offset).
  const uint32_t lds0 = (uint32_t)(uintptr_t)(&smem[wv][0][lane * 4]);
  const uint32_t lds1 = (uint32_t)(uintptr_t)(&smem[wv][1][lane * 4]);

  // Prime the pipeline: async-load row -> buffer 0.
  async_row_to_lds(x + (size_t)row * GROUP + lane * 4, lds0);

  int buf = 0;
  while (true) {
    const int  next     = row + nw;
    const bool has_next = next < nrows;

    if (has_next) {
      // Prefetch row+2*nw into L2 (depth-2 lookahead, no counter traffic).
      int far = next + nw;
      if (far < nrows)
        __builtin_prefetch(x + (size_t)far * GROUP + lane * 4, 0, 1);
      // Issue next row into the other buffer, then wait for the current one.
      async_row_to_lds(x + (size_t)next * GROUP + lane * 4, buf ? lds0 : lds1);
      asm volatile("s_wait_asynccnt 0x1" ::: "memory");   // current row landed
    } else {
      asm volatile("s_wait_asynccnt 0x0" ::: "memory");   // drain
    }

    // ---- consume current row from LDS (ds_load_b128) ----
    const v4f v = *(const v4f*)(&smem[wv][buf][lane * 4]);

    // v_min over the 128-element group
    float m = fminf(fminf(v.x, v.y), fminf(v.z, v.w));
    m = wave_min(m);

    const float cx = v.x - m, cy = v.y - m, cz = v.z - m, cw = v.w - m;

    // v_scale = ||x_centered||_2 / sqrt(128)
    float ss = cx * cx + cy * cy + cz * cz + cw * cw;
    ss = wave_sum(ss);
    const float vscale = sqrtf(ss) * 0.08838834764831845f;  // 1/sqrt(128)
    const float inv    = 1.0f / (vscale + 1e-10f);

    // LUT quantize/dequantize of unit values
    const float rx = lm_dequant(cx * inv);
    const float ry = lm_dequant(cy * inv);
    const float rz = lm_dequant(cz * inv);
    const float rw = lm_dequant(cw * inv);

    // refined gamma = <x_centered, recon> / (<recon, recon> + 1e-10)
    float num = cx * rx + cy * ry + cz * rz + cw * rw;
    float den = rx * rx + ry * ry + rz * rz + rw * rw;
#pragma unroll
    for (int off = 16; off; off >>= 1) {
      num += __shfl_xor(num, off, 32);
      den += __shfl_xor(den, off, 32);
    }
    const float gamma = num / (den + 1e-10f);

    v4f o;
    o.x = fmaf(rx, gamma, m);
    o.y = fmaf(ry, gamma, m);
    o.z = fmaf(rz, gamma, m);
    o.w = fmaf(rw, gamma, m);

    // Write-once stream: nontemporal b128 store.
    __builtin_nontemporal_store(o, (v4f*)(out + (size_t)row * GROUP + lane * 4));

    if (!has_next) break;
    row = next;
    buf ^= 1;
  }
}

extern "C" void kernel_launch(void* const* d_in, const int* in_sizes, int n_in,
                              void* d_out, int out_size, void* d_ws, size_t ws_size,
                              hipStream_t stream) {
  (void)n_in; (void)out_size; (void)d_ws; (void)ws_size;
  const float* x  = (const float*)d_in[0];
  float*      out = (float*)d_out;

  const long long n = (long long)in_sizes[0];
  const int nrows   = (int)(n / GROUP);
  if (nrows <= 0) return;

  int blocks = (nrows + WAVES_PER_BLOCK - 1) / WAVES_PER_BLOCK;
  if (blocks > 4096) blocks = 4096;   // 32K waves, grid-stride over rows

  hipLaunchKernelGGL(turboquant_fwd, dim3(blocks), dim3(256), 0, stream,
                     x, out, nrows);
}